// NMSLayer_34368328302798
// MI455X (gfx1250) — compile-verified
//
#include <hip/hip_runtime.h>
#include <hip/hip_bf16.h>
#include <stdint.h>

// Problem constants (from reference): x[64, 4096, 85]
#define NIMG       64
#define NMS_N      4096
#define BN_TOTAL   (NIMG * NMS_N)
#define ATTR       85
#define NCLS       80
#define NMS_T      1024               // threads per NMS block (32 waves, wave32)
#define NMS_IT     (NMS_N / NMS_T)    // boxes per thread = 4
#define CONF_TH    0.5f
#define SCORE_TH   0.3f
#define NMS_TH     0.4f
#define CLS_OFF    10.0f
#define NEG_KEY    (-1e30f)

typedef unsigned int v4u __attribute__((ext_vector_type(4)));
typedef int          v8i __attribute__((ext_vector_type(8)));
typedef int          v4i __attribute__((ext_vector_type(4)));

// ---------------------------------------------------------------------------
// Kernel 1: per-box preprocessing (bandwidth-bound streaming pass).
// Writes: det rows [batch,x1,y1,x2,y2,obj,maxscore,maxidx] to d_out,
//         shifted boxes / areas / sort keys to workspace.
// ---------------------------------------------------------------------------
__global__ __launch_bounds__(256)
void yolo_prep_kernel(const float* __restrict__ x,
                      float* __restrict__ key_ws,
                      float* __restrict__ box_ws,   // shifted corners, 4 per box
                      float* __restrict__ area_ws,
                      float* __restrict__ dets)
{
    const int gi = blockIdx.x * blockDim.x + threadIdx.x;
    if (gi >= BN_TOTAL) return;
    const float* __restrict__ r = x + (size_t)gi * ATTR;

    // gfx1250: global_prefetch_b8 for the tile the next block will stream.
    if (gi + 256 < BN_TOTAL) __builtin_prefetch(r + (size_t)256 * ATTR, 0, 0);

    float obj = r[4];
    const bool  pass = obj > CONF_TH;          // conf mask
    const float cm   = pass ? 1.0f : 0.0f;
    obj *= cm;

    const float px = r[0] * cm, py = r[1] * cm;
    const float pw = r[2] * cm, ph = r[3] * cm;
    const float x1 = px - pw * 0.5f, y1 = py - ph * 0.5f;
    const float x2 = px + pw * 0.5f, y2 = py + ph * 0.5f;

    // max / first-argmax over 80 masked class scores
    float ms = r[5] * cm;
    int   mi = 0;
#pragma unroll 8
    for (int j = 1; j < NCLS; ++j) {
        const float v = r[5 + j] * cm;
        if (v > ms) { ms = v; mi = j; }
    }

    const bool valid = pass && (ms > SCORE_TH);

    // class-offset shifted box; area computed from shifted corners exactly
    // like the reference (shift then subtract).
    const float off = (float)mi * CLS_OFF;
    const float sx1 = x1 + off, sy1 = y1 + off, sx2 = x2 + off, sy2 = y2 + off;

    ((float4*)box_ws)[gi] = make_float4(sx1, sy1, sx2, sy2);
    area_ws[gi] = (sx2 - sx1) * (sy2 - sy1);
    key_ws[gi]  = valid ? obj : NEG_KEY;

    const float bcol = (float)(gi >> 12);   // gi / 4096
    ((float4*)dets)[(size_t)gi * 2 + 0] = make_float4(bcol, x1, y1, x2);
    ((float4*)dets)[(size_t)gi * 2 + 1] = make_float4(y2, obj, ms, (float)mi);
}

// ---------------------------------------------------------------------------
// Kernel 2: per-image greedy NMS. One block per image, 1024 threads (32 waves).
// Keys DMA'd into LDS via the Tensor Data Mover; bitonic sort in LDS;
// serial greedy loop with parallel IoU suppression (boxes live in VGPRs).
// ---------------------------------------------------------------------------
__global__ __launch_bounds__(NMS_T)
void yolo_nms_kernel(const float* __restrict__ key_ws,
                     const float* __restrict__ box_ws,
                     const float* __restrict__ area_ws,
                     float* __restrict__ dets,
                     float* __restrict__ keep)
{
    __shared__ float          s_key[NMS_N];        // 16 KB
    __shared__ unsigned short s_idx[NMS_N];        // 8 KB
    __shared__ unsigned       s_sup[NMS_N / 32];   // suppression bitmask
    __shared__ float          s_bb[5];             // broadcast box + area
    __shared__ int            s_bidx;
    __shared__ int            s_pos;

    const int    tid  = threadIdx.x;
    const size_t base = (size_t)blockIdx.x * NMS_N;

    // ---- load this image's 4096 sort keys into LDS -------------------------
#if defined(__gfx1250__) && __has_builtin(__builtin_amdgcn_tensor_load_to_lds)
    if (tid < 32) {   // wave 0 issues one TDM descriptor (EXEC-independent DMA)
        const unsigned long long ga = (unsigned long long)(uintptr_t)(key_ws + base);
        const unsigned lds_addr = (unsigned)(uintptr_t)(&s_key[0]); // low 32b = LDS offset

        v4u g0;
        g0.x = 1u;                                             // count=1 (valid), user mode
        g0.y = lds_addr;                                       // lds_addr [63:32]
        g0.z = (unsigned)(ga & 0xFFFFFFFFull);                 // global_addr [95:64]
        g0.w = (unsigned)((ga >> 32) & 0x1FFFFFFull)           // global_addr [120:96]
             | (2u << 30);                                     // type=2 ("image")

        v8i g1;
        g1[0] = (int)(2u << 16);                 // workgroup_mask=0, data_size=2 (4B)
        g1[1] = (int)((NMS_N & 0xFFFF) << 16);   // tensor_dim0 low16
        g1[2] = (int)(1u << 16);                 // tensor_dim0 hi=0 | tensor_dim1=1
        g1[3] = (int)((unsigned)NMS_N << 16);    // tensor_dim1 hi=0 | tile_dim0=4096
        g1[4] = 1;                               // tile_dim1=1, tile_dim2=0
        g1[5] = NMS_N;                           // tensor_dim0_stride low32
        g1[6] = 0;                               // stride0 hi | stride1 low
        g1[7] = 0;

        const v4i gz4 = {0, 0, 0, 0};            // groups 2/3 unused (<=2D tensor)
        const v8i gz8 = {0, 0, 0, 0, 0, 0, 0, 0};
        __builtin_amdgcn_tensor_load_to_lds(g0, g1, gz4, gz4, gz8, 0);
        __builtin_amdgcn_s_wait_tensorcnt(0);
    }
#else
    for (int t = 0; t < NMS_IT; ++t) {
        const int i = tid + t * NMS_T;
        s_key[i] = key_ws[base + i];
    }
#endif
    if (tid < NMS_N / 32) s_sup[tid] = 0u;
    if (tid == 0) s_pos = 0;
    __syncthreads();

    // ---- per-thread register-resident boxes; mark invalid boxes suppressed -
    float rx1[NMS_IT], ry1[NMS_IT], rx2[NMS_IT], ry2[NMS_IT], ra[NMS_IT];
    bool  kept[NMS_IT];
#pragma unroll
    for (int t = 0; t < NMS_IT; ++t) {
        const int i = tid + t * NMS_T;
        const float4 bb = ((const float4*)box_ws)[base + i];
        rx1[t] = bb.x; ry1[t] = bb.y; rx2[t] = bb.z; ry2[t] = bb.w;
        ra[t]  = area_ws[base + i];
        kept[t] = false;
        s_idx[i] = (unsigned short)i;
        if (s_key[i] < -1e29f)                       // !valid -> pre-suppressed
            atomicOr(&s_sup[i >> 5], 1u << (i & 31));
    }

    // ---- bitonic sort of (key, idx) descending in LDS ----------------------
    for (unsigned k = 2; k <= NMS_N; k <<= 1) {
        for (unsigned j = k >> 1; j > 0; j >>= 1) {
            __syncthreads();
#pragma unroll
            for (int t = 0; t < NMS_IT; ++t) {
                const unsigned i = (unsigned)tid + (unsigned)t * NMS_T;
                const unsigned p = i ^ j;
                if (p > i) {
                    const float a = s_key[i], c = s_key[p];
                    const bool sw = ((i & k) == 0) ? (a < c) : (a > c);
                    if (sw) {
                        s_key[i] = c; s_key[p] = a;
                        const unsigned short ti = s_idx[i];
                        s_idx[i] = s_idx[p]; s_idx[p] = ti;
                    }
                }
            }
        }
    }

    // ---- greedy suppression loop -------------------------------------------
    for (;;) {
        __syncthreads();                 // prior-round sup bits visible to leader
        if (tid == 0) {
            int p = s_pos, found = -1;
            while (p < NMS_N) {
                const int oi = s_idx[p];
                if (((s_sup[oi >> 5] >> (oi & 31)) & 1u) == 0u) { found = oi; break; }
                ++p;
            }
            s_pos  = p + 1;
            s_bidx = found;
            if (found >= 0) {
                const float4 bb = ((const float4*)box_ws)[base + found];
                s_bb[0] = bb.x; s_bb[1] = bb.y; s_bb[2] = bb.z; s_bb[3] = bb.w;
                s_bb[4] = area_ws[base + found];
            }
        }
        __syncthreads();
        const int bidx = s_bidx;
        if (bidx < 0) break;

        const float bx1 = s_bb[0], by1 = s_bb[1], bx2 = s_bb[2], by2 = s_bb[3];
        const float ba  = s_bb[4];
#pragma unroll
        for (int t = 0; t < NMS_IT; ++t) {
            const int i = tid + t * NMS_T;
            if (i == bidx) kept[t] = true;
            const float ix1 = fmaxf(bx1, rx1[t]);
            const float iy1 = fmaxf(by1, ry1[t]);
            const float ix2 = fminf(bx2, rx2[t]);
            const float iy2 = fminf(by2, ry2[t]);
            const float inter = fmaxf(ix2 - ix1, 0.0f) * fmaxf(iy2 - iy1, 0.0f);
            const float iou   = inter / (ba + ra[t] - inter + 1e-9f);
            if (iou >= NMS_TH)
                atomicOr(&s_sup[i >> 5], 1u << (i & 31));
        }
    }

    // ---- outputs: keep flags, zero non-kept det rows -----------------------
#pragma unroll
    for (int t = 0; t < NMS_IT; ++t) {
        const int    i  = tid + t * NMS_T;
        const size_t gi = base + i;
        keep[gi] = kept[t] ? 1.0f : 0.0f;
        if (!kept[t]) {
            const float4 z = make_float4(0.f, 0.f, 0.f, 0.f);
            ((float4*)dets)[gi * 2 + 0] = z;
            ((float4*)dets)[gi * 2 + 1] = z;
        }
    }
}

// ---------------------------------------------------------------------------
extern "C" void kernel_launch(void* const* d_in, const int* in_sizes, int n_in,
                              void* d_out, int out_size, void* d_ws, size_t ws_size,
                              hipStream_t stream)
{
    const float* x = (const float*)d_in[0];

    float* dets = (float*)d_out;                        // [B*N*8]
    float* keep = dets + (size_t)BN_TOTAL * 8;          // [B*N] (bool -> 0/1 float)

    // workspace layout (floats): key[BN] | box[BN*4] | area[BN]  (~6.3 MB)
    float* wf      = (float*)d_ws;
    float* key_ws  = wf;
    float* box_ws  = wf + (size_t)BN_TOTAL;
    float* area_ws = wf + (size_t)BN_TOTAL * 5;

    yolo_prep_kernel<<<BN_TOTAL / 256, 256, 0, stream>>>(x, key_ws, box_ws, area_ws, dets);
    yolo_nms_kernel<<<NIMG, NMS_T, 0, stream>>>(key_ws, box_ws, area_ws, dets, keep);
}